// Subtraction2_14061722927494
// MI455X (gfx1250) — compile-verified
//
#include <hip/hip_runtime.h>
#include <stdint.h>

typedef float v4f __attribute__((ext_vector_type(4)));

#define PLANE_W   56
#define PLANE_HW  3136      // 56*56
#define NTAP      49        // 7*7

// One workgroup per (b,c) plane. Stage in1/in2 planes into LDS with async
// global->LDS B128 copies, then compute all 49 shifted subtractions with
// register-resident reflect-index tables and non-temporal float4 stores.
__global__ __launch_bounds__(256) void subtraction2_kernel(
    const float* __restrict__ in1,
    const float* __restrict__ in2,
    float* __restrict__ out)
{
    __shared__ __align__(16) float lds1[PLANE_HW];   // centers
    __shared__ __align__(16) float lds2[PLANE_HW];   // window source

    const int bc  = blockIdx.x;
    const int tid = threadIdx.x;

    const float* p1 = in1 + (size_t)bc * PLANE_HW;
    const float* p2 = in2 + (size_t)bc * PLANE_HW;

    // ---- Stage both planes to LDS: 784 async 16B transfers per plane ----
    for (int q = tid; q < PLANE_HW / 4; q += 256) {
        unsigned l1 = (unsigned)(uintptr_t)(&lds1[q * 4]);  // low 32b of flat = LDS offset
        unsigned l2 = (unsigned)(uintptr_t)(&lds2[q * 4]);
        const float* g1 = p1 + q * 4;
        const float* g2 = p2 + q * 4;
        asm volatile("global_load_async_to_lds_b128 %0, %1, off"
                     :: "v"(l1), "v"(g1) : "memory");
        asm volatile("global_load_async_to_lds_b128 %0, %1, off"
                     :: "v"(l2), "v"(g2) : "memory");
    }
    asm volatile("s_wait_asynccnt 0x0" ::: "memory");
    __syncthreads();

    float* outBase = out + (size_t)bc * NTAP * PLANE_HW;

    // ---- Compute: each thread owns float4 pixel groups (56%4==0 -> no row crossing) ----
    for (int g = tid; g < PLANE_HW / 4; g += 256) {
        const int l  = g * 4;
        const int y  = l / PLANE_W;
        const int x0 = l - y * PLANE_W;

        const v4f c4 = *(const v4f*)(&lds1[l]);

        // Reflect index tables (jnp 'reflect': -1->1, H->H-2  =>  p<0: -p ; p>=56: 110-p)
        int ry[7];
#pragma unroll
        for (int i = 0; i < 7; ++i) {
            int p = y + i - 3;
            ry[i] = (p < 0 ? -p : (p >= PLANE_W ? 110 - p : p)) * PLANE_W;
        }
        int rx[10];
#pragma unroll
        for (int q = 0; q < 10; ++q) {
            int p = x0 + q - 3;
            rx[q] = (p < 0 ? -p : (p >= PLANE_W ? 110 - p : p));
        }

#pragma unroll
        for (int i = 0; i < 7; ++i) {
            const float* row = lds2 + ry[i];
#pragma unroll
            for (int j = 0; j < 7; ++j) {
                v4f pv;
                pv.x = row[rx[j + 0]];
                pv.y = row[rx[j + 1]];
                pv.z = row[rx[j + 2]];
                pv.w = row[rx[j + 3]];
                const v4f o = c4 - pv;
                float* dst = outBase + (size_t)(i * 7 + j) * PLANE_HW + l;
                __builtin_nontemporal_store(o, (v4f*)dst);
            }
        }
    }
}

extern "C" void kernel_launch(void* const* d_in, const int* in_sizes, int n_in,
                              void* d_out, int out_size, void* d_ws, size_t ws_size,
                              hipStream_t stream)
{
    const float* in1 = (const float*)d_in[0];
    const float* in2 = (const float*)d_in[1];
    float* out = (float*)d_out;

    const int n_planes = in_sizes[0] / PLANE_HW;   // B*C = 512
    subtraction2_kernel<<<dim3(n_planes), dim3(256), 0, stream>>>(in1, in2, out);
}